// HyperPatchInvertedResidual_32375463477387
// MI455X (gfx1250) — compile-verified
//
#include <hip/hip_runtime.h>

typedef __attribute__((ext_vector_type(2))) float v2f;
typedef __attribute__((ext_vector_type(8))) float v8f;

#define HIDDEN 384
#define CIN    64
#define COUT   64
#define HW     128
#define NPAR   52608          // hidden*Cin + hidden*9 + Cout*hidden
#define N1     24576          // hidden*Cin
#define N2     3456           // hidden*9
#define PATCHES 256           // 16x16
#define EPSB   1e-5f

__device__ __forceinline__ int refl(int i, int n) {
    // jnp.pad mode='reflect' for +/-1 overhang: -1 -> 1, n -> n-2
    if (i < 0) i = -i;
    if (i >= n) i = 2 * n - 2 - i;
    return i;
}

__device__ __forceinline__ v8f wmma_f32(v2f a, v2f b, v8f c) {
    return __builtin_amdgcn_wmma_f32_16x16x4_f32(
        /*neg_a=*/false, a, /*neg_b=*/false, b,
        /*c_mod=*/(short)0, c, /*reuse_a=*/false, /*reuse_b=*/false);
}

// ---------------------------------------------------------------------------
// Kernel 1: per-patch pointwise 1x1 (W1 @ x) + BN1 + ReLU6 -> h1 (workspace)
// One block = one (b, fy, fx) patch. 256 threads = 8 waves.
// ---------------------------------------------------------------------------
__global__ __launch_bounds__(256) void k1_pw1(
    const float* __restrict__ x, const float* __restrict__ w,
    const float* __restrict__ g1, const float* __restrict__ b1,
    const float* __restrict__ m1, const float* __restrict__ v1,
    float* __restrict__ h1)
{
    __shared__ float xT[64 * 66];       // [pix][c], pad 66 -> conflict-free, 8B aligned
    __shared__ float w1s[128 * 66];     // [o_local][c]
    __shared__ float inv1[HIDDEN], sh1[HIDDEN];

    const int patch = blockIdx.x;
    const int b  = blockIdx.y;
    const int fy = patch >> 4, fx = patch & 15;
    const int y0 = fy * 8,  x0 = fx * 8;
    const int t  = threadIdx.x;
    const int lane = t & 31, wv = t >> 5;
    const int lm   = lane & 15;
    const int off  = (lane >> 4) << 1;           // 0 or 2 (K half per lane-half)
    const int ohi  = (lane >> 4) << 3;           // 0 or 8 (C/D row offset)

    // folded BN1 params
    for (int i = t; i < HIDDEN; i += 256) {
        float iv = g1[i] * rsqrtf(v1[i] + EPSB);
        inv1[i] = iv;
        sh1[i]  = b1[i] - m1[i] * iv;
    }
    // x patch, transposed to [pix][c]
    for (int idx = t; idx < 64 * 64; idx += 256) {
        int c = idx >> 6, pix = idx & 63;
        int py = pix >> 3, px = pix & 7;
        xT[pix * 66 + c] = x[((b * CIN + c) * HW + y0 + py) * HW + x0 + px];
    }

    const int wbase = b * NPAR * 256 + fy * 16 + fx;   // &w[b, p, fy, fx], p-stride 256

    for (int chunk = 0; chunk < 3; ++chunk) {
        __syncthreads();   // xT/bn ready (it 0); previous chunk's GEMM done (it >0)
        // load W1 rows [chunk*128, chunk*128+128), element p = o*64 + c
        const int pbase = wbase + chunk * 128 * 64 * 256;
        for (int idx = t; idx < 128 * 64; idx += 256) {
            int o = idx >> 6, c = idx & 63;
            w1s[o * 66 + c] = w[pbase + idx * 256];
        }
        __syncthreads();

        v8f acc[4];
        const v8f z8 = {0.f, 0.f, 0.f, 0.f, 0.f, 0.f, 0.f, 0.f};
#pragma unroll
        for (int nt = 0; nt < 4; ++nt) acc[nt] = z8;

        const float* arow = &w1s[(wv * 16 + lm) * 66];
#pragma unroll 4
        for (int kk = 0; kk < 16; ++kk) {
            v2f a = *(const v2f*)&arow[kk * 4 + off];
#pragma unroll
            for (int nt = 0; nt < 4; ++nt) {
                v2f bb = *(const v2f*)&xT[(nt * 16 + lm) * 66 + kk * 4 + off];
                acc[nt] = wmma_f32(a, bb, acc[nt]);
            }
        }
        // BN1 + ReLU6, scatter to h1[b][o][y][x]
        const int obase = chunk * 128 + wv * 16;
#pragma unroll
        for (int nt = 0; nt < 4; ++nt) {
            int pix = nt * 16 + lm;
            int py = pix >> 3, px = pix & 7;
#pragma unroll
            for (int i = 0; i < 8; ++i) {
                int o = obase + i + ohi;
                float v = acc[nt][i] * inv1[o] + sh1[o];
                v = fminf(fmaxf(v, 0.f), 6.f);
                h1[((b * HIDDEN + o) * HW + y0 + py) * HW + x0 + px] = v;
            }
        }
    }
}

// ---------------------------------------------------------------------------
// Kernel 2: depthwise 3x3 (reflect halo) + BN2 + ReLU6, then W3 GEMM + BN3
// + residual. Channel groups of 96 to bound LDS (~97 KB).
// ---------------------------------------------------------------------------
__global__ __launch_bounds__(256) void k2_dw_pw2(
    const float* __restrict__ x, const float* __restrict__ w,
    const float* __restrict__ h1,
    const float* __restrict__ g2, const float* __restrict__ b2,
    const float* __restrict__ m2, const float* __restrict__ v2,
    const float* __restrict__ g3, const float* __restrict__ b3,
    const float* __restrict__ m3, const float* __restrict__ v3,
    float* __restrict__ out)
{
    __shared__ float hh[96 * 100];      // halo window [ch][10*10]
    __shared__ float k2s[96 * 12];      // 3x3 taps, row pad 12
    __shared__ float w3s[64 * 98];      // [o][ch_local], pad 98
    __shared__ float h2T[64 * 98];      // [pix][ch_local], pad 98
    __shared__ float inv2[HIDDEN], sh2[HIDDEN], inv3[COUT], sh3[COUT];

    const int patch = blockIdx.x;
    const int b  = blockIdx.y;
    const int fy = patch >> 4, fx = patch & 15;
    const int y0 = fy * 8,  x0 = fx * 8;
    const int t  = threadIdx.x;
    const int lane = t & 31, wv = t >> 5;
    const int lm   = lane & 15;
    const int off  = (lane >> 4) << 1;
    const int ohi  = (lane >> 4) << 3;

    for (int i = t; i < HIDDEN; i += 256) {
        float iv = g2[i] * rsqrtf(v2[i] + EPSB);
        inv2[i] = iv;
        sh2[i]  = b2[i] - m2[i] * iv;
    }
    if (t < COUT) {
        float iv = g3[t] * rsqrtf(v3[t] + EPSB);
        inv3[t] = iv;
        sh3[t]  = b3[t] - m3[t] * iv;
    }

    // wave -> 2 of 16 output tiles; tiles 2w,2w+1 share mt = w>>1
    const int mt  = wv >> 1;
    const int nt0 = (wv & 1) << 1;      // nt0, nt0+1
    v8f acc[2];
    const v8f z8 = {0.f, 0.f, 0.f, 0.f, 0.f, 0.f, 0.f, 0.f};
    acc[0] = z8; acc[1] = z8;

    const int wbase = b * NPAR * 256 + fy * 16 + fx;

    for (int g = 0; g < 4; ++g) {
        __syncthreads();                 // previous group fully consumed
        const int ch0 = g * 96;
        // h1 halo window with reflect padding at map borders
        for (int idx = t; idx < 96 * 100; idx += 256) {
            int ch = idx / 100, pos = idx % 100;
            int py = pos / 10 - 1, px = pos % 10 - 1;
            int yy = refl(y0 + py, HW), xx = refl(x0 + px, HW);
            hh[idx] = h1[((b * HIDDEN + ch0 + ch) * HW + yy) * HW + xx];
        }
        // K2 taps: p = N1 + ch*9 + (dy*3+dx)
        for (int idx = t; idx < 96 * 9; idx += 256) {
            int ch = idx / 9, j = idx % 9;
            k2s[ch * 12 + j] = w[wbase + (N1 + (ch0 + ch) * 9 + j) * 256];
        }
        // W3 slice: p = N1 + N2 + o*hidden + ch
        for (int idx = t; idx < 64 * 96; idx += 256) {
            int o = idx / 96, c = idx % 96;
            w3s[o * 98 + c] = w[wbase + (N1 + N2 + o * HIDDEN + ch0 + c) * 256];
        }
        __syncthreads();

        // depthwise 3x3 + BN2 + ReLU6 -> h2T[pix][ch_local]
        for (int idx = t; idx < 96 * 64; idx += 256) {
            int ch = idx >> 6, pix = idx & 63;
            int py = pix >> 3, px = pix & 7;
            const float* hr = &hh[ch * 100 + py * 10 + px];
            const float* kk = &k2s[ch * 12];
            float s = 0.f;
#pragma unroll
            for (int dy = 0; dy < 3; ++dy)
#pragma unroll
                for (int dx = 0; dx < 3; ++dx)
                    s += hr[dy * 10 + dx] * kk[dy * 3 + dx];
            int chg = ch0 + ch;
            s = s * inv2[chg] + sh2[chg];
            s = fminf(fmaxf(s, 0.f), 6.f);
            h2T[pix * 98 + ch] = s;
        }
        __syncthreads();

        // GEMM partial: D[o,pix] += W3[o,ch] * h2[ch,pix], K = 96
        const float* arow = &w3s[(mt * 16 + lm) * 98];
#pragma unroll 4
        for (int kk = 0; kk < 24; ++kk) {
            v2f a = *(const v2f*)&arow[kk * 4 + off];
#pragma unroll
            for (int ti = 0; ti < 2; ++ti) {
                v2f bb = *(const v2f*)&h2T[((nt0 + ti) * 16 + lm) * 98 + kk * 4 + off];
                acc[ti] = wmma_f32(a, bb, acc[ti]);
            }
        }
    }

    // epilogue: BN3 + residual
#pragma unroll
    for (int ti = 0; ti < 2; ++ti) {
        int pix = (nt0 + ti) * 16 + lm;
        int py = pix >> 3, px = pix & 7;
#pragma unroll
        for (int i = 0; i < 8; ++i) {
            int o  = mt * 16 + i + ohi;
            int gi = ((b * COUT + o) * HW + y0 + py) * HW + x0 + px;
            out[gi] = acc[ti][i] * inv3[o] + sh3[o] + x[gi];
        }
    }
}

// ---------------------------------------------------------------------------
extern "C" void kernel_launch(void* const* d_in, const int* in_sizes, int n_in,
                              void* d_out, int out_size, void* d_ws, size_t ws_size,
                              hipStream_t stream) {
    (void)in_sizes; (void)n_in; (void)out_size; (void)ws_size;
    const float* x  = (const float*)d_in[0];
    const float* w  = (const float*)d_in[1];
    const float* g1 = (const float*)d_in[2];
    const float* b1 = (const float*)d_in[3];
    const float* m1 = (const float*)d_in[4];
    const float* v1 = (const float*)d_in[5];
    const float* g2 = (const float*)d_in[6];
    const float* b2 = (const float*)d_in[7];
    const float* m2 = (const float*)d_in[8];
    const float* v2 = (const float*)d_in[9];
    const float* g3 = (const float*)d_in[10];
    const float* b3 = (const float*)d_in[11];
    const float* m3 = (const float*)d_in[12];
    const float* v3 = (const float*)d_in[13];
    float* out = (float*)d_out;
    float* h1  = (float*)d_ws;   // 4*384*128*128 floats = 100.7 MB intermediate

    dim3 grid(PATCHES, 4);       // patch-fastest: per-batch 54MB weight set stays in 192MB L2
    dim3 blk(256);
    k1_pw1<<<grid, blk, 0, stream>>>(x, w, g1, b1, m1, v1, h1);
    k2_dw_pw2<<<grid, blk, 0, stream>>>(x, w, h1, g2, b2, m2, v2, g3, b3, m3, v3, out);
}